// MTP_11991548691074
// MI455X (gfx1250) — compile-verified
//
#include <hip/hip_runtime.h>
#include <hip/hip_bf16.h>
#include <math.h>

#define B_    2
#define T_    256
#define C_    1024
#define V_    32000
#define K_    4
#define M_    (T_ - K_)        // 252
#define NROWS (B_ * M_)        // 504
#define NPAD  512

typedef float v2f __attribute__((ext_vector_type(2)));
typedef float v8f __attribute__((ext_vector_type(8)));

// ---- CDNA5 async global->LDS path (ASYNCcnt), with safe fallback ----------
#if defined(__has_builtin)
#if __has_builtin(__builtin_amdgcn_global_load_async_to_lds_b128) && \
    __has_builtin(__builtin_amdgcn_s_wait_asynccnt)
#define USE_ASYNC_LDS 1
#endif
#endif

typedef int v4i_ __attribute__((vector_size(4 * sizeof(int))));
typedef __attribute__((address_space(1))) v4i_ gv4i_t;   // global int4
typedef __attribute__((address_space(3))) v4i_ lv4i_t;   // LDS int4

__device__ __forceinline__ void async_copy16(void* lds_dst, const void* gsrc) {
#ifdef USE_ASYNC_LDS
    __builtin_amdgcn_global_load_async_to_lds_b128(
        (gv4i_t*)(void*)gsrc, (lv4i_t*)lds_dst, 0, 0);
#else
    *(float4*)lds_dst = *(const float4*)gsrc;
#endif
}

__device__ __forceinline__ void wait_async_le3() {
#ifdef USE_ASYNC_LDS
    __builtin_amdgcn_s_wait_asynccnt(3);
#endif
}
__device__ __forceinline__ void wait_async_le0() {
#ifdef USE_ASYNC_LDS
    __builtin_amdgcn_s_wait_asynccnt(0);
#endif
}

// -------------------------------------------------------------------------
// prep: cat[row] = [ rmsnorm(emb[tok]*sqrt(C)) , rmsnorm(h_prev) ]
// -------------------------------------------------------------------------
__global__ __launch_bounds__(256) void mtp_prep_kernel(
    const float* __restrict__ hidden, const int* __restrict__ tokens,
    const float* __restrict__ emb, const float* __restrict__ curr_prev,
    float* __restrict__ cat, int k)
{
    __shared__ float red[256];
    const int row = blockIdx.x;
    const int tid = threadIdx.x;
    float* crow = cat + (size_t)row * (2 * C_);
    if (row >= NROWS) {
        for (int c = tid; c < 2 * C_; c += 256) crow[c] = 0.f;
        return;
    }
    const int b = row / M_;
    const int t = row % M_;
    const int tok = tokens[b * T_ + (k + 1 + t)];
    const float* e  = emb + (size_t)tok * C_;
    const float* hp = (k == 0) ? (hidden + ((size_t)b * T_ + t) * C_)
                               : (curr_prev + (size_t)row * C_);
    float s = 0.f;
    for (int c = tid; c < C_; c += 256) { float x = e[c]; s += x * x; }
    red[tid] = s; __syncthreads();
    for (int off = 128; off > 0; off >>= 1) {
        if (tid < off) red[tid] += red[tid + off];
        __syncthreads();
    }
    const float denom1 = 32.f * sqrtf(red[0] * (1.f / C_)) + 1e-8f;
    __syncthreads();
    s = 0.f;
    for (int c = tid; c < C_; c += 256) { float x = hp[c]; s += x * x; }
    red[tid] = s; __syncthreads();
    for (int off = 128; off > 0; off >>= 1) {
        if (tid < off) red[tid] += red[tid + off];
        __syncthreads();
    }
    const float denom2 = sqrtf(red[0] * (1.f / C_)) + 1e-8f;
    const float inv1 = 32.f / denom1;
    const float inv2 = 1.f / denom2;
    for (int c = tid; c < C_; c += 256) {
        crow[c]      = e[c]  * inv1;
        crow[C_ + c] = hp[c] * inv2;
    }
}

// -------------------------------------------------------------------------
// LayerNorm: xn = a*(x-mean)/(std+1e-6)+b
// -------------------------------------------------------------------------
__global__ __launch_bounds__(256) void mtp_ln_kernel(
    const float* __restrict__ x, const float* __restrict__ a,
    const float* __restrict__ bvec, float* __restrict__ xn)
{
    __shared__ float red[256];
    const int row = blockIdx.x;
    const int tid = threadIdx.x;
    const float* xr = x + (size_t)row * C_;
    float s = 0.f;
    for (int c = tid; c < C_; c += 256) s += xr[c];
    red[tid] = s; __syncthreads();
    for (int off = 128; off > 0; off >>= 1) {
        if (tid < off) red[tid] += red[tid + off];
        __syncthreads();
    }
    const float mean = red[0] * (1.f / C_);
    __syncthreads();
    s = 0.f;
    for (int c = tid; c < C_; c += 256) { float d = xr[c] - mean; s += d * d; }
    red[tid] = s; __syncthreads();
    for (int off = 128; off > 0; off >>= 1) {
        if (tid < off) red[tid] += red[tid + off];
        __syncthreads();
    }
    const float inv = 1.f / (sqrtf(red[0] * (1.f / C_)) + 1e-6f);
    float* xo = xn + (size_t)row * C_;
    for (int c = tid; c < C_; c += 256)
        xo[c] = a[c] * (xr[c] - mean) * inv + bvec[c];
}

// -------------------------------------------------------------------------
// f32 WMMA GEMM, 64x128 block tile, double-buffered async LDS staging.
// 8 waves: wave (wr = w&3, wc = w>>2) owns 16 rows x 64 cols (4 accums).
// modes: 0 store, 1 store to out+out2, 2 residual add, 3 GELU, 4 logits.
// -------------------------------------------------------------------------
#define BM 64
#define BN 128
#define BK 16
#define SA_LD (BK + 4)    // 20 floats: rows stay 16B aligned, banks spread
#define SB_LD (BN + 4)    // 132 floats

__global__ __launch_bounds__(256) void mtp_gemm_wmma(
    const float* __restrict__ A, int lda,
    const float* __restrict__ Bm, int ldb, int bcol0,
    const float* __restrict__ bias,
    float* __restrict__ out, int ldo,
    float* __restrict__ out2,
    int Kdim, int mode, int kidx)
{
    __shared__ float sA[2][BM][SA_LD];
    __shared__ float sB[2][BK][SB_LD];

    const int tid  = threadIdx.x;
    const int wave = tid >> 5;
    const int lane = tid & 31;
    const int wr = wave & 3;            // row slot (x16)
    const int wc = wave >> 2;           // col slot (x64)
    const int lh = lane >> 4;           // lane half
    const int ll = lane & 15;
    const int rowBase = blockIdx.y * BM;
    const int colBase = blockIdx.x * BN;

    v8f acc[4];
#pragma unroll
    for (int t = 0; t < 4; ++t) acc[t] = (v8f){0.f,0.f,0.f,0.f,0.f,0.f,0.f,0.f};

    // staging assignments (per thread, 3 x 16B per tile)
    const int ar  = tid >> 2,  ac = (tid & 3) << 2;   // A: 64 rows x 16 cols
    const int br0 = tid >> 5,  bc = (tid & 31) << 2;  // B: rows br0, br0+8 x 128 cols
    const float* Ag = A + (size_t)(rowBase + ar) * lda + ac;
    const float* Bg = Bm + (size_t)br0 * ldb + bcol0 + colBase + bc;
    const size_t ldb8 = (size_t)8 * ldb;

#define ISSUE_TILE(kk, buf)                                                   \
    do {                                                                      \
        async_copy16(&sA[buf][ar][ac], Ag + (kk));                            \
        async_copy16(&sB[buf][br0][bc],     Bg + (size_t)(kk) * ldb);         \
        async_copy16(&sB[buf][br0 + 8][bc], Bg + (size_t)(kk) * ldb + ldb8);  \
    } while (0)

    ISSUE_TILE(0, 0);
    int p = 0;
    for (int kk = 0; kk < Kdim; kk += BK) {
        const bool hasNext = (kk + BK) < Kdim;
        if (hasNext) { ISSUE_TILE(kk + BK, p ^ 1); wait_async_le3(); }
        else         { wait_async_le0(); }
        __syncthreads();

        const int mrow = wr * 16 + ll;
        const int nc0  = wc * 64 + ll;
#pragma unroll
        for (int s = 0; s < 4; ++s) {
            const int kb = s * 4 + lh * 2;   // lanes 0-15: K0/K1, 16-31: K2/K3
            v2f a; a[0] = sA[p][mrow][kb]; a[1] = sA[p][mrow][kb + 1];
#pragma unroll
            for (int t = 0; t < 4; ++t) {
                v2f b;
                b[0] = sB[p][kb][nc0 + t * 16];
                b[1] = sB[p][kb + 1][nc0 + t * 16];
                acc[t] = __builtin_amdgcn_wmma_f32_16x16x4_f32(
                    false, a, false, b, (short)0, acc[t], false, false);
            }
        }
        __syncthreads();
        p ^= 1;
    }

    // ---------------- epilogue ----------------
#pragma unroll
    for (int t = 0; t < 4; ++t) {
        const int col = colBase + wc * 64 + t * 16 + ll;
        const float bvl = bias ? bias[col] : 0.f;
#pragma unroll
        for (int vr = 0; vr < 8; ++vr) {
            const int row = rowBase + wr * 16 + vr + lh * 8;
            const float val = acc[t][vr] + bvl;
            if (mode == 0) {
                out[(size_t)row * ldo + col] = val;
            } else if (mode == 1) {
                out [(size_t)row * ldo + col] = val;
                out2[(size_t)row * ldo + col] = val;
            } else if (mode == 2) {
                out[(size_t)row * ldo + col] += val;
            } else if (mode == 3) {
                const float g = 0.5f * val *
                    (1.f + tanhf(0.7978845608028654f * (val + 0.044715f * val * val * val)));
                out[(size_t)row * ldo + col] = g;
            } else { // logits
                if (row < NROWS)
                    out[((size_t)row * K_ + kidx) * V_ + col] = val;
            }
        }
    }
#undef ISSUE_TILE
}

// -------------------------------------------------------------------------
extern "C" void kernel_launch(void* const* d_in, const int* in_sizes, int n_in,
                              void* d_out, int out_size, void* d_ws, size_t ws_size,
                              hipStream_t stream)
{
    (void)in_sizes; (void)n_in; (void)out_size; (void)ws_size;
    const float* hidden   = (const float*)d_in[0];
    const int*   tokens   = (const int*)  d_in[1];
    const float* proj_w   = (const float*)d_in[2];
    const float* proj_b   = (const float*)d_in[3];
    const float* ln1_a    = (const float*)d_in[4];
    const float* ln1_b    = (const float*)d_in[5];
    const float* attn_w   = (const float*)d_in[6];
    const float* attn_b   = (const float*)d_in[7];
    const float* attn_pw  = (const float*)d_in[8];
    const float* attn_pb  = (const float*)d_in[9];
    const float* ln2_a    = (const float*)d_in[10];
    const float* ln2_b    = (const float*)d_in[11];
    const float* fc_w     = (const float*)d_in[12];
    const float* fc_b     = (const float*)d_in[13];
    const float* fcp_w    = (const float*)d_in[14];
    const float* fcp_b    = (const float*)d_in[15];
    const float* unembd_w = (const float*)d_in[16];
    const float* unembd_b = (const float*)d_in[17];
    const float* emb      = (const float*)d_in[18];
    float* out = (float*)d_out;

    float* ws   = (float*)d_ws;
    float* cat  = ws;                                  // NPAD * 2C
    float* xbuf = cat  + (size_t)NPAD * 2 * C_;        // NPAD * C
    float* xn   = xbuf + (size_t)NPAD * C_;            // NPAD * C
    float* vbuf = xn   + (size_t)NPAD * C_;            // NPAD * C
    float* curr = vbuf + (size_t)NPAD * C_;            // NPAD * C
    float* hb   = curr + (size_t)NPAD * C_;            // NPAD * 4C

    const dim3 blk(256);
    const int rt = NPAD / BM;   // 8 row tiles

    for (int k = 0; k < K_; ++k) {
        mtp_prep_kernel<<<NPAD, blk, 0, stream>>>(hidden, tokens, emb, curr, cat, k);

        // curr = x = cat @ proj_w[k] + proj_b[k]
        mtp_gemm_wmma<<<dim3(C_ / BN, rt), blk, 0, stream>>>(
            cat, 2 * C_,
            proj_w + (size_t)k * 2 * C_ * C_, C_, 0,
            proj_b + (size_t)k * C_,
            curr, C_, xbuf, 2 * C_, /*mode=*/1, k);

        mtp_ln_kernel<<<NPAD, blk, 0, stream>>>(xbuf, ln1_a + (size_t)k * C_,
                                                ln1_b + (size_t)k * C_, xn);

        // v = xn @ attn_w[k][:, 2C:3C] + attn_b[k][2C:]
        mtp_gemm_wmma<<<dim3(C_ / BN, rt), blk, 0, stream>>>(
            xn, C_,
            attn_w + (size_t)k * C_ * 3 * C_, 3 * C_, 2 * C_,
            attn_b + (size_t)k * 3 * C_ + 2 * C_,
            vbuf, C_, nullptr, C_, /*mode=*/0, k);

        // x += v @ attn_pw[k] + attn_pb[k]
        mtp_gemm_wmma<<<dim3(C_ / BN, rt), blk, 0, stream>>>(
            vbuf, C_,
            attn_pw + (size_t)k * C_ * C_, C_, 0,
            attn_pb + (size_t)k * C_,
            xbuf, C_, nullptr, C_, /*mode=*/2, k);

        mtp_ln_kernel<<<NPAD, blk, 0, stream>>>(xbuf, ln2_a + (size_t)k * C_,
                                                ln2_b + (size_t)k * C_, xn);

        // h = gelu(xn @ fc_w[k] + fc_b[k])
        mtp_gemm_wmma<<<dim3(4 * C_ / BN, rt), blk, 0, stream>>>(
            xn, C_,
            fc_w + (size_t)k * C_ * 4 * C_, 4 * C_, 0,
            fc_b + (size_t)k * 4 * C_,
            hb, 4 * C_, nullptr, C_, /*mode=*/3, k);

        // x += h @ fcp_w[k] + fcp_b[k]
        mtp_gemm_wmma<<<dim3(C_ / BN, rt), blk, 0, stream>>>(
            hb, 4 * C_,
            fcp_w + (size_t)k * 4 * C_ * C_, C_, 0,
            fcp_b + (size_t)k * C_,
            xbuf, C_, nullptr, 4 * C_, /*mode=*/2, k);

        // logits[:, k, :] = x @ unembd_w + unembd_b
        mtp_gemm_wmma<<<dim3(V_ / BN, rt), blk, 0, stream>>>(
            xbuf, C_,
            unembd_w, V_, 0,
            unembd_b,
            out, V_, nullptr, C_, /*mode=*/4, k);
    }
}